// WarpV2_14877766714075
// MI455X (gfx1250) — compile-verified
//
#include <hip/hip_runtime.h>

// dense_image_warp (bilinear, TFA clamp semantics) for
//   img: [B=8, H=512, W=512, C=32] f32 (NHWC)
//   flo: [B, H, W, 2] f32 (dx, dy)
// out[b,y,x,c] = bilinear sample of img at (y + dy, x + dx) with
// floor indices clamped to [0, size-2] and alphas clamped to [0,1].
//
// Memory-bound gather kernel: 8 lanes per pixel, float4 per lane so every
// neighbor fetch is a coalesced global_load_b128 (128 B per pixel line).
// Output stored non-temporally (never re-read) to preserve L2 for the
// img gather working set.

typedef float  f4 __attribute__((ext_vector_type(4)));
typedef float  f2 __attribute__((ext_vector_type(2)));

#define DIW_B 8
#define DIW_H 512
#define DIW_W 512
#define DIW_C 32
// log2 helpers for the fixed power-of-two sizes
#define DIW_WSH 9    // log2(W)
#define DIW_HWSH 18  // log2(H*W)

__global__ __launch_bounds__(256) void
dense_image_warp_f32(const float* __restrict__ img,
                     const float* __restrict__ flo,
                     float* __restrict__ out,
                     int n_threads)
{
    int gtid = blockIdx.x * blockDim.x + threadIdx.x;
    if (gtid >= n_threads) return;

    // 8 lanes per pixel, 4 channels per lane
    const int pix = gtid >> 3;            // [0, B*H*W)
    const int c0  = (gtid & 7) << 2;      // channel start: 0,4,...,28

    const int x = pix & (DIW_W - 1);
    const int y = (pix >> DIW_WSH) & (DIW_H - 1);
    // b = pix >> DIW_HWSH  (folded into base below)

    // flow for this pixel (broadcast across the 8 lanes of the pixel)
    const f2 f = *(const f2*)(flo + ((long)pix << 1));
    const float qx = (float)x + f.x;
    const float qy = (float)y + f.y;

    // TFA _interpolate_bilinear clamping
    float x0f = fminf(fmaxf(floorf(qx), 0.0f), (float)(DIW_W - 2));
    float y0f = fminf(fmaxf(floorf(qy), 0.0f), (float)(DIW_H - 2));
    const float ax = fminf(fmaxf(qx - x0f, 0.0f), 1.0f);
    const float ay = fminf(fmaxf(qy - y0f, 0.0f), 1.0f);
    const int x0 = (int)x0f;
    const int y0 = (int)y0f;

    // element index of (b, y0, x0, c0); all indices < 2^27, fits int32
    const int bOff  = (pix >> DIW_HWSH) << (DIW_HWSH);       // b*H*W
    const int base  = ((bOff + (y0 << DIW_WSH) + x0) << 5) + c0;  // *C
    const int rowC  = DIW_W * DIW_C;                         // stride one row

    const f4 tl = *(const f4*)(img + base);
    const f4 tr = *(const f4*)(img + base + DIW_C);
    const f4 bl = *(const f4*)(img + base + rowC);
    const f4 br = *(const f4*)(img + base + rowC + DIW_C);

    // top = tl + ax*(tr-tl); bot = bl + ax*(br-bl); out = top + ay*(bot-top)
    f4 top, bot, o;
    top.x = fmaf(ax, tr.x - tl.x, tl.x);
    top.y = fmaf(ax, tr.y - tl.y, tl.y);
    top.z = fmaf(ax, tr.z - tl.z, tl.z);
    top.w = fmaf(ax, tr.w - tl.w, tl.w);
    bot.x = fmaf(ax, br.x - bl.x, bl.x);
    bot.y = fmaf(ax, br.y - bl.y, bl.y);
    bot.z = fmaf(ax, br.z - bl.z, bl.z);
    bot.w = fmaf(ax, br.w - bl.w, bl.w);
    o.x = fmaf(ay, bot.x - top.x, top.x);
    o.y = fmaf(ay, bot.y - top.y, top.y);
    o.z = fmaf(ay, bot.z - top.z, top.z);
    o.w = fmaf(ay, bot.w - top.w, top.w);

    // output never re-read: non-temporal b128 store keeps L2 for img gathers
    __builtin_nontemporal_store(o, (f4*)(out + ((long)pix << 5) + c0));
}

extern "C" void kernel_launch(void* const* d_in, const int* in_sizes, int n_in,
                              void* d_out, int out_size, void* d_ws, size_t ws_size,
                              hipStream_t stream)
{
    const float* img = (const float*)d_in[0];
    const float* flo = (const float*)d_in[1];
    float* out = (float*)d_out;

    // total threads: one per 4 output channels
    const int n_threads = DIW_B * DIW_H * DIW_W * (DIW_C / 4); // 16,777,216
    const int block = 256;
    const int grid = (n_threads + block - 1) / block;          // 65,536

    dense_image_warp_f32<<<grid, block, 0, stream>>>(img, flo, out, n_threads);
}